// ChannelShift_80075370267210
// MI455X (gfx1250) — compile-verified
//
#include <hip/hip_runtime.h>

// ChannelShift, M=1, strategy='start', n_div=8 on x:[64,512,56,56] f32.
// hw = 3136, fold = 392. In float4 units: HW4 = 784, FOLD4 = 98.
//   p4 in [0,  98): out[b,c,p4] = x[b,c+1,p4]   (zero when c == 511)
//   p4 in [98,196): out[b,c,p4] = x[b,c-1,p4]   (zero when c == 0)
//   p4 >= 196     : out[b,c,p4] = x[b,c,  p4]
// Pure streaming permutation: 822 MB total traffic, ~35 us floor @ 23.3 TB/s.
// Key structural fact: with 256 threads/block, only the first 256-float4 slice
// of a (b,c) row intersects the shifted regions; slices 1..3 are identity.

typedef float v4f __attribute__((ext_vector_type(4)));

#define CS_BS    64
#define CS_C     512
#define CS_HW4   784   // 3136 floats per (b,c) row = 784 float4
#define CS_FOLD4 98    // 392 floats = 98 float4

__global__ __launch_bounds__(256)
void channel_shift_f32_kernel(const v4f* __restrict__ in4, v4f* __restrict__ out4) {
    const int  c   = blockIdx.x;                                    // 0..511
    const long row = ((long)blockIdx.y * CS_C + c) * (long)CS_HW4;  // float4 idx of row start
    const int  tid = threadIdx.x;

    // ---- slice 0: p4 = tid in [0,256) — the only slice with shifted regions ----
    const int shift = (tid < CS_FOLD4) ? 1 : ((tid < 2 * CS_FOLD4) ? -1 : 0);
    const int srcC  = c + shift;
    v4f v0;
    v0.x = 0.0f; v0.y = 0.0f; v0.z = 0.0f; v0.w = 0.0f;
    if ((unsigned)srcC < (unsigned)CS_C)   // zero-pad rows fall out of this range
        v0 = __builtin_nontemporal_load(in4 + row + (long)shift * CS_HW4 + tid);

    // ---- slices 1,2: p4 in [256,768) — pure identity copy, no predication ----
    const v4f v1 = __builtin_nontemporal_load(in4 + row + 256 + tid);
    const v4f v2 = __builtin_nontemporal_load(in4 + row + 512 + tid);

    __builtin_nontemporal_store(v0, out4 + row + tid);
    __builtin_nontemporal_store(v1, out4 + row + 256 + tid);
    __builtin_nontemporal_store(v2, out4 + row + 512 + tid);

    // ---- tail: p4 in [768,784) — 16 float4s, identity ----
    if (tid < CS_HW4 - 768) {
        const v4f v3 = __builtin_nontemporal_load(in4 + row + 768 + tid);
        __builtin_nontemporal_store(v3, out4 + row + 768 + tid);
    }
}

extern "C" void kernel_launch(void* const* d_in, const int* in_sizes, int n_in,
                              void* d_out, int out_size, void* d_ws, size_t ws_size,
                              hipStream_t stream) {
    (void)in_sizes; (void)n_in; (void)d_ws; (void)ws_size; (void)out_size;

    const v4f* in4  = (const v4f*)d_in[0];
    v4f*       out4 = (v4f*)d_out;

    dim3 grid(CS_C, CS_BS);   // one block per (channel, batch) row
    dim3 block(256);          // 8 wave32s
    channel_shift_f32_kernel<<<grid, block, 0, stream>>>(in4, out4);
}